// Triton_8031588844327
// MI455X (gfx1250) — compile-verified
//
#include <hip/hip_runtime.h>

typedef __attribute__((ext_vector_type(2))) float v2f;
typedef __attribute__((ext_vector_type(8))) float v8f;

#define IN_F    1024
#define D_K     64
#define D_V     64
#define N_HEADS 16
#define SEQ_L   32768
#define QK_STRIDE 1032   // padded LDS row stride (floats) to spread LDS banks

// workspace layout (float offsets)
#define WS_QK     0
#define WS_CQ     16384
#define WS_LOGITS 16512
#define WS_PART   (16512 + N_HEADS * SEQ_L)          // 8 split-K partials of ctx
#define WS_CTX    (WS_PART + 8 * N_HEADS * IN_F)

// ---------------------------------------------------------------------------
// Kernel A: q[h,k] = Wq[h]·ut + bq;  qk[h,f] = sum_k q[h,k]*Wk[h,k,f];
//           cq[h] = q[h]·bk[h]
// ---------------------------------------------------------------------------
__global__ void proj_q_kernel(const float* __restrict__ ut,
                              const float* __restrict__ Wq, const float* __restrict__ bq,
                              const float* __restrict__ Wk, const float* __restrict__ bk,
                              float* __restrict__ qk, float* __restrict__ cq)
{
    __shared__ float s_ut[IN_F];
    __shared__ float s_q[N_HEADS * D_K];
    for (int i = threadIdx.x; i < IN_F; i += blockDim.x) s_ut[i] = ut[i];
    __syncthreads();
    for (int o = threadIdx.x; o < N_HEADS * D_K; o += blockDim.x) {
        const float* w = Wq + (size_t)o * IN_F;
        float acc = bq[o];
        for (int f = 0; f < IN_F; ++f) acc = fmaf(s_ut[f], w[f], acc);
        s_q[o] = acc;
    }
    __syncthreads();
    for (int o = threadIdx.x; o < N_HEADS * IN_F; o += blockDim.x) {
        int h = o >> 10, f = o & (IN_F - 1);
        const float* w  = Wk + (size_t)h * D_K * IN_F + f;
        const float* qh = s_q + h * D_K;
        float acc = 0.f;
        for (int k = 0; k < D_K; ++k) acc = fmaf(qh[k], w[(size_t)k * IN_F], acc);
        qk[o] = acc;
    }
    if (threadIdx.x < N_HEADS) {
        int h = threadIdx.x;
        float acc = 0.f;
        for (int k = 0; k < D_K; ++k) acc = fmaf(s_q[h * D_K + k], bk[h * D_K + k], acc);
        cq[h] = acc;
    }
}

// ---------------------------------------------------------------------------
// Kernel B: logits[h,l] = (kt[l,:]·qk[h,:] + cq[h]) * 0.125
// One wave per 16-row tile of kt; WMMA f32 16x16x4, K=1024.
// A = kt tile (16xK), B = qk^T (Kx16 heads), D[m][h] -> logits[h][row+m].
// ---------------------------------------------------------------------------
__global__ void logits_kernel(const float* __restrict__ kt,
                              const float* __restrict__ qk, const float* __restrict__ cq,
                              float* __restrict__ logits)
{
    __shared__ float s_qk[N_HEADS * QK_STRIDE];
    __shared__ float s_cq[N_HEADS];
    for (int i = threadIdx.x; i < N_HEADS * IN_F; i += blockDim.x) {
        int h = i >> 10, f = i & (IN_F - 1);
        s_qk[h * QK_STRIDE + f] = qk[i];
    }
    if (threadIdx.x < N_HEADS) s_cq[threadIdx.x] = cq[threadIdx.x];
    __syncthreads();

    const int lane  = threadIdx.x & 31;
    const int wave  = threadIdx.x >> 5;
    const int n     = lane & 15;            // A row within tile / B column (head)
    const int khalf = (lane >> 4) << 1;     // lanes 16-31 hold K+2,K+3
    const int nTiles = SEQ_L / 16;          // 2048
    const int wavesTotal = (gridDim.x * blockDim.x) >> 5;

    for (int tile = blockIdx.x * (blockDim.x >> 5) + wave; tile < nTiles; tile += wavesTotal) {
        const int rowBase  = tile << 4;
        const float* aPtr  = kt + (size_t)(rowBase + n) * IN_F + khalf;
        const float* bPtr  = s_qk + n * QK_STRIDE + khalf;
        v8f acc = {};
        for (int kk = 0; kk < IN_F; kk += 4) {
            v2f a = *(const v2f*)(aPtr + kk);                 // global_load_b64
            v2f b = *(const v2f*)(bPtr + kk);                 // ds_load_b64
            acc = __builtin_amdgcn_wmma_f32_16x16x4_f32(false, a, false, b,
                                                        (short)0, acc, false, false);
        }
        const float cqh = s_cq[n];          // epilogue: this lane's head == n
        const int   mOff = (lane >> 4) << 3;
        float* dst = logits + (size_t)n * SEQ_L + rowBase + mOff;
        #pragma unroll
        for (int j = 0; j < 8; ++j)
            dst[j] = (acc[j] + cqh) * 0.125f;   // 1/sqrt(64)
    }
}

// ---------------------------------------------------------------------------
// Kernel C: softmax over L per head (in place). One block per head.
// ---------------------------------------------------------------------------
__global__ void softmax_kernel(float* __restrict__ logits)
{
    __shared__ float s_red[256];
    float* row = logits + (size_t)blockIdx.x * SEQ_L;
    const int tid = threadIdx.x;

    float m = -3.402823466e38f;
    for (int i = tid; i < SEQ_L; i += 256) m = fmaxf(m, row[i]);
    s_red[tid] = m; __syncthreads();
    for (int s = 128; s > 0; s >>= 1) {
        if (tid < s) s_red[tid] = fmaxf(s_red[tid], s_red[tid + s]);
        __syncthreads();
    }
    const float gmax = s_red[0];
    __syncthreads();

    float sum = 0.f;
    for (int i = tid; i < SEQ_L; i += 256) sum += __expf(row[i] - gmax);
    s_red[tid] = sum; __syncthreads();
    for (int s = 128; s > 0; s >>= 1) {
        if (tid < s) s_red[tid] += s_red[tid + s];
        __syncthreads();
    }
    const float inv = 1.f / s_red[0];
    for (int i = tid; i < SEQ_L; i += 256)
        row[i] = __expf(row[i] - gmax) * inv;
}

// ---------------------------------------------------------------------------
// Kernel D: ctx partials. ctx[h,f] = sum_l attn[h,l]*vt[l,f].
// 64 blocks = 64 column tiles of 16; 8 waves/block = split-K (4096 each).
// A = attn (16 heads x K), B = vt (K x 16 cols). Deterministic: each wave
// writes its own partial buffer; fixed-order reduction follows.
// ---------------------------------------------------------------------------
__global__ void ctx_kernel(const float* __restrict__ attn, const float* __restrict__ vt,
                           float* __restrict__ partial)
{
    const int lane  = threadIdx.x & 31;
    const int wave  = threadIdx.x >> 5;      // split-K index 0..7
    const int n0    = blockIdx.x << 4;       // column tile base
    const int n     = lane & 15;
    const int khalf = (lane >> 4) << 1;
    const int KC    = SEQ_L / 8;             // 4096
    const int k0    = wave * KC;

    const float* aPtr = attn + (size_t)n * SEQ_L + k0 + khalf;
    v8f acc = {};
    for (int kk = 0; kk < KC; kk += 4) {
        v2f a = *(const v2f*)(aPtr + kk);                       // attn hits L2
        const size_t krow = (size_t)(k0 + kk + khalf);
        v2f b;
        b.x = vt[krow * IN_F + n0 + n];                         // streaming vt
        b.y = vt[(krow + 1) * IN_F + n0 + n];
        acc = __builtin_amdgcn_wmma_f32_16x16x4_f32(false, a, false, b,
                                                    (short)0, acc, false, false);
    }
    const int hOff = (lane >> 4) << 3;       // D: VGPR j -> head j or j+8
    float* dst = partial + (size_t)wave * (N_HEADS * IN_F);
    #pragma unroll
    for (int j = 0; j < 8; ++j)
        dst[(size_t)(j + hOff) * IN_F + n0 + n] = acc[j];
}

__global__ void ctx_reduce_kernel(const float* __restrict__ partial, float* __restrict__ ctx)
{
    int i = blockIdx.x * blockDim.x + threadIdx.x;
    if (i < N_HEADS * IN_F) {
        float s = 0.f;
        #pragma unroll
        for (int p = 0; p < 8; ++p) s += partial[(size_t)p * (N_HEADS * IN_F) + i];
        ctx[i] = s;
    }
}

// ---------------------------------------------------------------------------
// Kernel E: mh[t] = Wv[t,:]·ctx[h(t),:] + bv[t], then LayerNorm over 1024.
// ---------------------------------------------------------------------------
__global__ void out_kernel(const float* __restrict__ Wv, const float* __restrict__ bv,
                           const float* __restrict__ ctx,
                           const float* __restrict__ gamma, const float* __restrict__ beta,
                           float* __restrict__ out)
{
    __shared__ float s_red[IN_F];
    const int t = threadIdx.x;               // 0..1023 == h*64 + dv
    const int h = t >> 6;
    const float* w = Wv + (size_t)t * IN_F;
    const float* c = ctx + (size_t)h * IN_F;
    float acc = bv[t];
    for (int f = 0; f < IN_F; ++f) acc = fmaf(w[f], c[f], acc);

    s_red[t] = acc; __syncthreads();
    for (int s = 512; s > 0; s >>= 1) {
        if (t < s) s_red[t] += s_red[t + s];
        __syncthreads();
    }
    const float mu = s_red[0] * (1.f / IN_F);
    __syncthreads();
    const float d = acc - mu;
    s_red[t] = d * d; __syncthreads();
    for (int s = 512; s > 0; s >>= 1) {
        if (t < s) s_red[t] += s_red[t + s];
        __syncthreads();
    }
    const float var = s_red[0] * (1.f / IN_F);
    out[t] = gamma[t] * d * rsqrtf(var + 1e-5f) + beta[t];
}

// ---------------------------------------------------------------------------
extern "C" void kernel_launch(void* const* d_in, const int* in_sizes, int n_in,
                              void* d_out, int out_size, void* d_ws, size_t ws_size,
                              hipStream_t stream)
{
    const float* ut    = (const float*)d_in[0];
    const float* kt    = (const float*)d_in[1];
    const float* vt    = (const float*)d_in[2];
    const float* Wq    = (const float*)d_in[3];
    const float* bq    = (const float*)d_in[4];
    const float* Wk    = (const float*)d_in[5];
    const float* bk    = (const float*)d_in[6];
    const float* Wv    = (const float*)d_in[7];
    const float* bv    = (const float*)d_in[8];
    const float* gamma = (const float*)d_in[9];
    const float* beta  = (const float*)d_in[10];

    float* ws      = (float*)d_ws;
    float* qk      = ws + WS_QK;
    float* cq      = ws + WS_CQ;
    float* logits  = ws + WS_LOGITS;
    float* partial = ws + WS_PART;
    float* ctx     = ws + WS_CTX;
    float* out     = (float*)d_out;

    proj_q_kernel  <<<1,       256,  0, stream>>>(ut, Wq, bq, Wk, bk, qk, cq);
    logits_kernel  <<<256,     256,  0, stream>>>(kt, qk, cq, logits);
    softmax_kernel <<<N_HEADS, 256,  0, stream>>>(logits);
    ctx_kernel     <<<64,      256,  0, stream>>>(logits, vt, partial);
    ctx_reduce_kernel<<<64,    256,  0, stream>>>(partial, ctx);
    out_kernel     <<<1,       1024, 0, stream>>>(Wv, bv, ctx, gamma, beta, out);
}